// ite_GCN_42365557407791
// MI455X (gfx1250) — compile-verified
//
#include <hip/hip_runtime.h>
#include <hip/hip_bf16.h>

// ---------------------------------------------------------------------------
// GCN inference on MI455X (gfx1250, wave32).
//   per iter: support = h @ Wgc  (WMMA f32 16x16x4, W pre-paired in LDS)
//             h' = relu(0.5*h + 0.5*(A @ support + b))   (CSR pull, no atomics)
//   final:   logits = h @ Wlin_pad (WMMA), log_softmax (wave shuffle reduce)
// ---------------------------------------------------------------------------

typedef __attribute__((ext_vector_type(2))) float v2f;
typedef __attribute__((ext_vector_type(8))) float v8f;

#define DFEAT 96
#define NCLS  40
#define NPAD  48
#define NITER 8

// ---------------- small utility kernels ----------------

__global__ void zero_i32_k(int* p, int n) {
    int i = blockIdx.x * blockDim.x + threadIdx.x;
    if (i < n) p[i] = 0;
}

__global__ void copy_f32_k(const float* __restrict__ src, float* __restrict__ dst, int n) {
    int i = blockIdx.x * blockDim.x + threadIdx.x;
    if (i < n) dst[i] = src[i];
}

// W_lin [96 x 40] -> [96 x 48] zero padded
__global__ void pad_wlin_k(const float* __restrict__ w, float* __restrict__ wp) {
    int i = blockIdx.x * blockDim.x + threadIdx.x;
    if (i >= DFEAT * NPAD) return;
    int k = i / NPAD, c = i % NPAD;
    wp[i] = (c < NCLS) ? w[k * NCLS + c] : 0.0f;
}

// ---------------- CSR build ----------------

__global__ void hist_k(const int* __restrict__ row, int* __restrict__ cnt, int e) {
    int i = blockIdx.x * blockDim.x + threadIdx.x;
    if (i < e) atomicAdd(&cnt[row[i]], 1);
}

// single-block exclusive scan (1024 threads, Hillis-Steele per chunk + carry)
__global__ void scan_k(const int* __restrict__ cnt, int* __restrict__ ptr, int n) {
    __shared__ int tmp[1024];
    __shared__ int carry;
    int t = threadIdx.x;
    if (t == 0) carry = 0;
    __syncthreads();
    for (int base = 0; base < n; base += 1024) {
        int v = (base + t < n) ? cnt[base + t] : 0;
        tmp[t] = v;
        __syncthreads();
        for (int off = 1; off < 1024; off <<= 1) {
            int add = (t >= off) ? tmp[t - off] : 0;
            __syncthreads();
            tmp[t] += add;
            __syncthreads();
        }
        int incl = tmp[t];
        if (base + t < n) ptr[base + t] = carry + (incl - v);
        __syncthreads();
        if (t == 1023) carry += incl;
        __syncthreads();
    }
    if (t == 0) ptr[n] = carry;
}

__global__ void scatter_k(const int* __restrict__ erow, const int* __restrict__ ecol,
                          const float* __restrict__ eval, const int* __restrict__ ptr,
                          int* __restrict__ fill, int* __restrict__ ccol,
                          float* __restrict__ cval, int e) {
    int i = blockIdx.x * blockDim.x + threadIdx.x;
    if (i >= e) return;
    int r = erow[i];
    int idx = ptr[r] + atomicAdd(&fill[r], 1);
    ccol[idx] = ecol[i];
    cval[idx] = eval[i];
}

// ---------------- WMMA GEMM: Out[nrows x NCOLS] = A[nrows x 96] @ W[96 x NCOLS] ----
// 256 threads = 8 waves; each wave owns TWO 16-row M-subtiles (32 rows) so every
// B fragment (one ds_load_b64 from LDS, K-pair-swizzled + bank-pad) feeds 2 WMMAs.
// FP32 WMMA 16x16x4 layouts (ISA 7.12.2):
//   A frag v2f: lane m=l&15 holds K = 2*(l>>4)+{0,1}
//   B frag v2f: lane n=l&15 holds rows K = 2*(l>>4)+{0,1} of column n0+n
//   C frag v8f: vgpr r holds rows r (lanes 0-15) / r+8 (lanes 16-31)

template <int NCOLS>
__global__ void __launch_bounds__(256)
gemm_wmma_k(const float* __restrict__ A, const float* __restrict__ W,
            float* __restrict__ Out, int nrows) {
    constexpr int NT   = NCOLS / 16;
    // need 2*LSTR % 64 == 32 dwords so lane halves hit disjoint bank halves
    constexpr int LSTR = (NCOLS % 32 == 16) ? NCOLS : NCOLS + 16;
    __shared__ v2f w2[(DFEAT / 2) * LSTR];
    for (int i = threadIdx.x; i < (DFEAT / 2) * NCOLS; i += 256) {
        int kk = i / NCOLS, n = i % NCOLS;
        v2f t;
        t.x = W[(2 * kk)     * NCOLS + n];
        t.y = W[(2 * kk + 1) * NCOLS + n];
        w2[kk * LSTR + n] = t;
    }
    __syncthreads();

    const int wave = threadIdx.x >> 5;
    const int lane = threadIdx.x & 31;
    const int sup0 = blockIdx.x * 8 + wave;      // 32-row super-tile index
    const int row0 = sup0 * 32;
    if (row0 >= nrows) return;
    const bool has2 = (row0 + 32) <= nrows;      // second 16-row subtile valid?
    const int row1 = has2 ? (row0 + 16) : row0;  // clamp -> duplicate work, masked store
    const int m  = lane & 15;
    const int kh = lane >> 4;                    // K offset = 2*kh
    const float* ar0 = A + (size_t)(row0 + m) * DFEAT;
    const float* ar1 = A + (size_t)(row1 + m) * DFEAT;

    v8f c0[NT] = {};
    v8f c1[NT] = {};
#pragma unroll
    for (int ks = 0; ks < DFEAT / 4; ++ks) {
        const int k0 = ks * 4 + 2 * kh;
        v2f a0 = *(const v2f*)(ar0 + k0);        // 8B aligned (k0 even, DFEAT even)
        v2f a1 = *(const v2f*)(ar1 + k0);
        const v2f* wrow = w2 + (ks * 2 + kh) * LSTR;
#pragma unroll
        for (int t = 0; t < NT; ++t) {
            v2f b = wrow[t * 16 + m];            // single ds_load_b64, no re-pairing
            c0[t] = __builtin_amdgcn_wmma_f32_16x16x4_f32(
                false, a0, false, b, (short)0, c0[t], false, false);
            c1[t] = __builtin_amdgcn_wmma_f32_16x16x4_f32(
                false, a1, false, b, (short)0, c1[t], false, false);
        }
    }
#pragma unroll
    for (int t = 0; t < NT; ++t) {
        const int n = t * 16 + m;
#pragma unroll
        for (int r = 0; r < 8; ++r) {
            Out[(size_t)(row0 + r + 8 * kh) * NCOLS + n] = c0[t][r];
            if (has2)
                Out[(size_t)(row1 + r + 8 * kh) * NCOLS + n] = c1[t][r];
        }
    }
}

// ---------------- aggregation + skip + ReLU ----------------
// one wave32 per node, lane owns features {lane, lane+32, lane+64}

__global__ void __launch_bounds__(128)
agg_relu_k(const float* __restrict__ h, const float* __restrict__ sup,
           const int* __restrict__ ptr, const int* __restrict__ col,
           const float* __restrict__ val, const float* __restrict__ bias,
           float* __restrict__ hout, int n) {
    const int wave = threadIdx.x >> 5;
    const int lane = threadIdx.x & 31;
    const int node = blockIdx.x * 4 + wave;
    if (node >= n) return;
    const int e0 = ptr[node], e1 = ptr[node + 1];
    float a0 = 0.f, a1 = 0.f, a2 = 0.f;
    for (int e = e0; e < e1; ++e) {
        const int cc = col[e];            // wave-uniform -> scalar load
        const float v = val[e];
        const float* s = sup + (size_t)cc * DFEAT;
        a0 = fmaf(v, s[lane],      a0);
        a1 = fmaf(v, s[lane + 32], a1);
        a2 = fmaf(v, s[lane + 64], a2);
    }
    const size_t base = (size_t)node * DFEAT;
    float o;
    o = 0.5f * h[base + lane]      + 0.5f * (a0 + bias[lane]);
    hout[base + lane]      = fmaxf(o, 0.f);
    o = 0.5f * h[base + lane + 32] + 0.5f * (a1 + bias[lane + 32]);
    hout[base + lane + 32] = fmaxf(o, 0.f);
    o = 0.5f * h[base + lane + 64] + 0.5f * (a2 + bias[lane + 64]);
    hout[base + lane + 64] = fmaxf(o, 0.f);
}

// ---------------- log-softmax over 40 classes (logits padded to 48/row) -------

__global__ void __launch_bounds__(256)
logsoftmax_k(const float* __restrict__ logits, float* __restrict__ out, int n) {
    const int wave = threadIdx.x >> 5;
    const int lane = threadIdx.x & 31;
    const int node = blockIdx.x * 8 + wave;
    if (node >= n) return;
    const float* lr = logits + (size_t)node * NPAD;
    float v0 = lr[lane];                              // c = lane (0..31)
    float v1 = (lane < 8) ? lr[32 + lane] : -3.4e38f; // c = 32+lane (32..39)
    float mx = fmaxf(v0, v1);
#pragma unroll
    for (int off = 16; off; off >>= 1) mx = fmaxf(mx, __shfl_xor(mx, off, 32));
    float s = __expf(v0 - mx) + ((lane < 8) ? __expf(v1 - mx) : 0.f);
#pragma unroll
    for (int off = 16; off; off >>= 1) s += __shfl_xor(s, off, 32);
    const float lse = mx + __logf(s);
    out[(size_t)node * NCLS + lane] = v0 - lse;
    if (lane < 8) out[(size_t)node * NCLS + 32 + lane] = v1 - lse;
}

// ---------------- launcher ----------------

extern "C" void kernel_launch(void* const* d_in, const int* in_sizes, int n_in,
                              void* d_out, int out_size, void* d_ws, size_t ws_size,
                              hipStream_t stream) {
    const float* x    = (const float*)d_in[0];
    const int*   erow = (const int*)  d_in[1];
    const int*   ecol = (const int*)  d_in[2];
    const float* eval = (const float*)d_in[3];
    const float* Wgc  = (const float*)d_in[4];
    const float* bgc  = (const float*)d_in[5];
    const float* Wlin = (const float*)d_in[6];
    float* out = (float*)d_out;

    const int N = in_sizes[0] / DFEAT;
    const int E = in_sizes[1];

    // carve workspace
    char* p = (char*)d_ws;
    auto carve = [&](size_t bytes) {
        void* r = (void*)p;
        p += (bytes + 255) & ~(size_t)255;
        return r;
    };
    float* hA   = (float*)carve((size_t)N * DFEAT * 4);
    float* hB   = (float*)carve((size_t)N * DFEAT * 4);
    float* sup  = (float*)carve((size_t)N * DFEAT * 4);  // reused as logits [N x 48]
    float* wpad = (float*)carve((size_t)DFEAT * NPAD * 4);
    int*   ptr  = (int*)  carve((size_t)(N + 1) * 4);
    int*   cnt  = (int*)  carve((size_t)N * 4);
    int*   ccol = (int*)  carve((size_t)E * 4);
    float* cval = (float*)carve((size_t)E * 4);
    if ((size_t)(p - (char*)d_ws) > ws_size) return;  // insufficient scratch

    const int TB = 256;
    // ---- CSR build ----
    zero_i32_k<<<(N + TB - 1) / TB, TB, 0, stream>>>(cnt, N);
    hist_k<<<(E + TB - 1) / TB, TB, 0, stream>>>(erow, cnt, E);
    scan_k<<<1, 1024, 0, stream>>>(cnt, ptr, N);
    zero_i32_k<<<(N + TB - 1) / TB, TB, 0, stream>>>(cnt, N);
    scatter_k<<<(E + TB - 1) / TB, TB, 0, stream>>>(erow, ecol, eval, ptr, cnt,
                                                    ccol, cval, E);
    // ---- init h, pad W_lin ----
    copy_f32_k<<<((N * DFEAT) + TB - 1) / TB, TB, 0, stream>>>(x, hA, N * DFEAT);
    pad_wlin_k<<<(DFEAT * NPAD + TB - 1) / TB, TB, 0, stream>>>(Wlin, wpad);

    const int supTiles   = (N + 31) / 32;
    const int gemmBlocks = (supTiles + 7) / 8;
    const int aggBlocks  = (N + 3) / 4;

    float* cur = hA;
    float* nxt = hB;
    for (int it = 0; it < NITER; ++it) {
        gemm_wmma_k<DFEAT><<<gemmBlocks, 256, 0, stream>>>(cur, Wgc, sup, N);
        agg_relu_k<<<aggBlocks, 128, 0, stream>>>(cur, sup, ptr, ccol, cval,
                                                  bgc, nxt, N);
        float* t = cur; cur = nxt; nxt = t;
    }
    // ---- classifier + log-softmax ----
    gemm_wmma_k<NPAD><<<gemmBlocks, 256, 0, stream>>>(cur, wpad, sup, N);
    logsoftmax_k<<<(N + 7) / 8, 256, 0, stream>>>(sup, out, N);
}